// MissHitScatter_31980326486572
// MI455X (gfx1250) — compile-verified
//
#include <hip/hip_runtime.h>

// MissHitScatter (miss_hit_gate + scatter, IS_HIT=true, PATH_NUM=4):
//   out[0]    = inputs   (256 MiB copy)
//   out[1..3] = 0        (768 MiB zero-fill)
// Pure streaming op: 0 FLOPs, ~1.25 GiB HBM traffic -> ~55us floor @ 23.3 TB/s.
// CDNA5 path: global_load_b128/global_store_b128, NT temporal hints (touch-once,
// don't churn the 192MB L2), saddr + 32-bit voffset (scale_offset) addressing.
// Each block owns a tile of 1024 consecutive float4 chunks; thread t touches
// tile_base + t + {0,256,512,768}, so the 4 loads/stores per iteration differ
// only in a compile-time immediate (0/4096/8192/12288 bytes) and can be
// clause'd with 4 cachelines in flight per wave.

typedef float floatx4 __attribute__((ext_vector_type(4)));

#define THREADS 256u
#define UNROLL  4u

__global__ void __launch_bounds__(THREADS)
copy_path0_kernel(const floatx4* __restrict__ in, floatx4* __restrict__ out,
                  unsigned int n4) {
    const unsigned int stride = gridDim.x * (THREADS * UNROLL);
    unsigned int base = blockIdx.x * (THREADS * UNROLL) + threadIdx.x;

    // Main body: 4 NT loads (constant immediates), then 4 NT stores.
    for (; base + 3u * THREADS < n4; base += stride) {
        floatx4 v0 = __builtin_nontemporal_load(in + base);
        floatx4 v1 = __builtin_nontemporal_load(in + base + THREADS);
        floatx4 v2 = __builtin_nontemporal_load(in + base + 2u * THREADS);
        floatx4 v3 = __builtin_nontemporal_load(in + base + 3u * THREADS);
        __builtin_nontemporal_store(v0, out + base);
        __builtin_nontemporal_store(v1, out + base + THREADS);
        __builtin_nontemporal_store(v2, out + base + 2u * THREADS);
        __builtin_nontemporal_store(v3, out + base + 3u * THREADS);
    }
    // Tail (never taken for the 2^24-chunk problem size; kept for generality).
    for (; base < n4; base += THREADS) {
        floatx4 v = __builtin_nontemporal_load(in + base);
        __builtin_nontemporal_store(v, out + base);
    }
}

__global__ void __launch_bounds__(THREADS)
zero_paths_kernel(floatx4* __restrict__ out, unsigned int n4) {
    const unsigned int stride = gridDim.x * (THREADS * UNROLL);
    const floatx4 z = {0.0f, 0.0f, 0.0f, 0.0f};
    unsigned int base = blockIdx.x * (THREADS * UNROLL) + threadIdx.x;

    for (; base + 3u * THREADS < n4; base += stride) {
        __builtin_nontemporal_store(z, out + base);
        __builtin_nontemporal_store(z, out + base + THREADS);
        __builtin_nontemporal_store(z, out + base + 2u * THREADS);
        __builtin_nontemporal_store(z, out + base + 3u * THREADS);
    }
    for (; base < n4; base += THREADS) {
        __builtin_nontemporal_store(z, out + base);
    }
}

extern "C" void kernel_launch(void* const* d_in, const int* in_sizes, int n_in,
                              void* d_out, int out_size, void* d_ws, size_t ws_size,
                              hipStream_t stream) {
    const float* in = (const float*)d_in[0];
    float* out = (float*)d_out;

    const unsigned int n_in_elems = (unsigned int)in_sizes[0];           // 65536*1024
    const unsigned int n4_copy    = n_in_elems / 4u;                     // 2^24 chunks
    const unsigned int n_zero     = (unsigned int)out_size - n_in_elems; // paths 1..3
    const unsigned int n4_zero    = n_zero / 4u;                         // 3*2^24 chunks

    const unsigned int chunks_per_thread = 32u;  // 8 iterations of the x4 body

    unsigned int blocks_c = (n4_copy + THREADS * chunks_per_thread - 1u) /
                            (THREADS * chunks_per_thread);
    unsigned int blocks_z = (n4_zero + THREADS * chunks_per_thread - 1u) /
                            (THREADS * chunks_per_thread);
    if (blocks_c < 1u) blocks_c = 1u;
    if (blocks_z < 1u) blocks_z = 1u;

    copy_path0_kernel<<<dim3(blocks_c), dim3(THREADS), 0, stream>>>(
        (const floatx4*)in, (floatx4*)out, n4_copy);

    zero_paths_kernel<<<dim3(blocks_z), dim3(THREADS), 0, stream>>>(
        (floatx4*)(out + n_in_elems), n4_zero);
}